// VectorQuantizer_13013750907262
// MI455X (gfx1250) — compile-verified
//
#include <hip/hip_runtime.h>

typedef float v2f __attribute__((ext_vector_type(2)));
typedef float v8f __attribute__((ext_vector_type(8)));

#define NUM_EMB 512
#define EMB_DIM 64
#define TDIM    4096
#define BDIM    64
#define NTOK    (BDIM * TDIM)            // 262144 tokens
#define NTOT    (BDIM * EMB_DIM * TDIM)  // 16777216 output elements
#define LSTRIDE 66                       // padded LDS row stride (floats)

__global__ void vq_zero_losses(float* out) {
    out[NTOT]     = 0.0f;
    out[NTOT + 1] = 0.0f;
}

__global__ __launch_bounds__(256)
void vq_main(const float* __restrict__ z, const float* __restrict__ emb,
             float* __restrict__ out) {
    extern __shared__ float smem[];
    float* ldsEmb  = smem;                         // 512 * 66 floats
    float* ldsE2   = ldsEmb + NUM_EMB * LSTRIDE;   // 512 floats
    int*   ldsIdx  = (int*)(ldsE2 + NUM_EMB);      // 128 ints
    float* ldsPart = (float*)(ldsIdx + 128);       // 8 floats

    const int tid  = threadIdx.x;
    const int tau0 = blockIdx.x * 128;             // first token of this WG
    const int b    = tau0 >> 12;                   // tau0 / TDIM
    const int t0   = tau0 & (TDIM - 1);
    const float* zb = z + b * (EMB_DIM * TDIM);

    // ---- stage codebook into LDS (row-major, stride 66 => conflict-free B frags)
    for (int i = tid; i < NUM_EMB * EMB_DIM / 2; i += 256) {
        int n = i >> 5;                 // 32 float2 per codebook row
        int c = (i & 31) * 2;
        v2f v = *(const v2f*)(emb + n * EMB_DIM + c);
        *(v2f*)(&ldsEmb[n * LSTRIDE + c]) = v;
    }
    __syncthreads();

    // ---- precompute ||e_n||^2
    for (int n = tid; n < NUM_EMB; n += 256) {
        float s = 0.0f;
        #pragma unroll 16
        for (int d = 0; d < EMB_DIM; ++d) {
            float e = ldsEmb[n * LSTRIDE + d];
            s += e * e;
        }
        ldsE2[n] = s;
    }
    __syncthreads();

    // ---- phase 1: WMMA distance GEMM + argmin (each wave owns 16 tokens)
    const int lane = tid & 31;
    const int wv   = tid >> 5;
    const int col  = lane & 15;     // M for A / N for B-C / token within wave-tile
    const int hi   = lane >> 4;     // 0: K=0,1 half; 1: K=2,3 half
    const int tw   = t0 + wv * 16;

    // A fragments: 16x64 of z, 16 steps of 16x4 (f32 A layout per ISA 7.12.2)
    v2f a2[16];
    #pragma unroll
    for (int s = 0; s < 16; ++s) {
        int d0 = 4 * s + 2 * hi;
        a2[s].x = zb[d0 * TDIM + tw + col];
        a2[s].y = zb[(d0 + 1) * TDIM + tw + col];
    }

    float bestv[8];
    int   besti[8];
    #pragma unroll
    for (int v = 0; v < 8; ++v) { bestv[v] = 3.4e38f; besti[v] = 0x7fffffff; }

    for (int tile = 0; tile < 32; ++tile) {
        int n = tile * 16 + col;
        const float* bp = &ldsEmb[n * LSTRIDE + 2 * hi];
        v2f bf[16];
        #pragma unroll
        for (int s = 0; s < 16; ++s) bf[s] = *(const v2f*)(bp + 4 * s);

        v8f acc = {0.f, 0.f, 0.f, 0.f, 0.f, 0.f, 0.f, 0.f};
        #pragma unroll
        for (int s = 0; s < 16; ++s) {
            // D = A(16x4) * B(4x16) + C, exact f32
            acc = __builtin_amdgcn_wmma_f32_16x16x4_f32(
                false, a2[s], false, bf[s], (short)0, acc, false, false);
        }

        float e2n = ldsE2[n];
        #pragma unroll
        for (int v = 0; v < 8; ++v) {
            float dv = e2n - 2.0f * acc[v];   // ||e||^2 - 2 z.e (argmin-equivalent)
            if (dv < bestv[v] || (dv == bestv[v] && n < besti[v])) {
                bestv[v] = dv; besti[v] = n;
            }
        }
    }

    // cross-lane argmin within each 16-lane half (rows 0-7 / 8-15)
    #pragma unroll
    for (int off = 8; off >= 1; off >>= 1) {
        #pragma unroll
        for (int v = 0; v < 8; ++v) {
            float ov = __shfl_xor(bestv[v], off, 32);
            int   oi = __shfl_xor(besti[v], off, 32);
            if (ov < bestv[v] || (ov == bestv[v] && oi < besti[v])) {
                bestv[v] = ov; besti[v] = oi;
            }
        }
    }
    if (col == 0) {
        #pragma unroll
        for (int v = 0; v < 8; ++v) ldsIdx[wv * 16 + 8 * hi + v] = besti[v];
    }
    __syncthreads();

    // ---- phase 2: gather + coalesced store + loss accumulation
    float local = 0.0f;
    #pragma unroll 4
    for (int e = tid; e < 128 * EMB_DIM; e += 256) {
        int d = e >> 7;          // dim
        int j = e & 127;         // token within WG (consecutive t => coalesced)
        int ii = ldsIdx[j];
        float qv = ldsEmb[ii * LSTRIDE + d];
        float zv = zb[d * TDIM + t0 + j];
        out[(b * EMB_DIM + d) * TDIM + t0 + j] = qv;
        float df = qv - zv;
        local += df * df;
    }

    #pragma unroll
    for (int off = 16; off >= 1; off >>= 1) local += __shfl_xor(local, off, 32);
    if (lane == 0) ldsPart[wv] = local;
    __syncthreads();
    if (tid == 0) {
        float s = 0.0f;
        #pragma unroll
        for (int w = 0; w < 8; ++w) s += ldsPart[w];
        const float inv = 1.0f / (float)NTOT;
        atomicAdd(&out[NTOT],     s * inv);           // vq_loss
        atomicAdd(&out[NTOT + 1], s * (0.25f * inv)); // commitment_loss
    }
}

extern "C" void kernel_launch(void* const* d_in, const int* in_sizes, int n_in,
                              void* d_out, int out_size, void* d_ws, size_t ws_size,
                              hipStream_t stream) {
    const float* z   = (const float*)d_in[0];
    const float* emb = (const float*)d_in[1];
    float* out = (float*)d_out;

    size_t lds_bytes = (size_t)(NUM_EMB * LSTRIDE + NUM_EMB) * sizeof(float)
                     + 128 * sizeof(int) + 8 * sizeof(float);

    hipLaunchKernelGGL(vq_zero_losses, dim3(1), dim3(1), 0, stream, out);
    hipLaunchKernelGGL(vq_main, dim3(NTOK / 128), dim3(256), lds_bytes, stream,
                       z, emb, out);
}